// ConcatenationAggregator_16758962389079
// MI455X (gfx1250) — compile-verified
//
#include <hip/hip_runtime.h>
#include <stdint.h>

typedef __attribute__((ext_vector_type(2))) float v2f;
typedef __attribute__((ext_vector_type(8))) float v8f;

#define D_ENT    64
#define D_OUT_   128
#define TILE_M   128
#define A_STRIDE 196                     // 192 + 4 pad -> conflict-free frag reads
#define KSTEPS   48                      // 192 / 4
#define NTILES   8                       // 128 / 16

#define A_LDS_FLOATS (TILE_M * A_STRIDE)        // 25088 floats / buffer
#define W_LDS_FLOATS (KSTEPS * NTILES * 32 * 2) // 24576 floats
#define LDS_BYTES    ((2 * A_LDS_FLOATS + W_LDS_FLOATS) * 4)  // 299008 B <= 320KB/WGP

// Async copy: each lane copies 16 bytes from the global address in vaddr to the LDS address in vdst, tracked by ASYNCcnt.
__device__ __forceinline__ void async_b128(unsigned lds_off, const void* g) {
    unsigned long long ga = (unsigned long long)(uintptr_t)g;
    asm volatile("global_load_async_to_lds_b128 %0, %1, off"
                 :: "v"(lds_off), "v"(ga) : "memory");
}

__device__ __forceinline__ void wait_async0() {
    asm volatile("s_wait_asynccnt 0x0" ::: "memory");
}

// out[r, :] = relu( [review[r], user[adj_u[r]], item[adj_i[r]]] @ W )
// (reference's fixed column perms of user/item == fixed row-perm of W; folded.)
__global__ void __launch_bounds__(256)
concat_agg_gemm(const float* __restrict__ review,
                const float* __restrict__ user,
                const float* __restrict__ item,
                const float* __restrict__ W,
                const int*   __restrict__ adj_u,
                const int*   __restrict__ adj_i,
                float* __restrict__ out,
                int n_rows, int n_tiles)
{
    extern __shared__ float smem[];
    float* Wlds = smem + 2 * A_LDS_FLOATS;   // [48][8][32][2] B-fragments

    const int tid  = threadIdx.x;
    const int lane = tid & 31;
    const int wave = tid >> 5;

    // ---- Stage W into fragment-ready layout (once per block) ----
    // frag (ks, nt): lane l holds B[k0 + 2*(l>>4) + {0,1}][nt*16 + (l&15)]
    for (int f = tid; f < KSTEPS * NTILES * 32; f += 256) {
        int l  = f & 31;
        int nt = (f >> 5) & 7;
        int ks = f >> 8;
        int k  = ks * 4 + ((l >> 4) << 1);
        int n  = nt * 16 + (l & 15);
        v2f w2;
        w2.x = W[k * D_OUT_ + n];
        w2.y = W[(k + 1) * D_OUT_ + n];
        *(v2f*)(&Wlds[f * 2]) = w2;
    }

    const int lp     = tid & 15;   // 16 lanes cover one 256B row segment
    const int rowSub = tid >> 4;   // 0..15
    // Low 32 bits of a generic __shared__ pointer == LDS byte offset (aperture rule).
    const unsigned ldsA0 = (unsigned)(uintptr_t)(void*)smem;

    // Issue the async gather of one 128-row A tile into buffer `buf`.
    auto issue_gather = [&](long rowBase, int buf) {
        const unsigned base = ldsA0 + (unsigned)buf * (A_LDS_FLOATS * 4);
        #pragma unroll
        for (int pass = 0; pass < 8; ++pass) {
            int  rIC = pass * 16 + rowSub;                  // 0..127
            long r   = rowBase + rIC;
            long rc  = (r < n_rows) ? r : (long)(n_rows - 1);
            unsigned dst = base + (unsigned)((rIC * A_STRIDE + lp * 4) * 4);
            async_b128(dst,       review + rc * D_ENT + lp * 4);
            long ui = (long)adj_u[rc];
            async_b128(dst + 256, user + ui * D_ENT + lp * 4);
            long ii = (long)adj_i[rc];
            async_b128(dst + 512, item + ii * D_ENT + lp * 4);
        }
    };

    // Prologue: start gather of this block's first tile into buffer 0.
    issue_gather((long)blockIdx.x * TILE_M, 0);
    int buf = 0;

    for (int tile = blockIdx.x; tile < n_tiles; tile += gridDim.x) {
        wait_async0();        // tile's A data landed in LDS (and Wlds on 1st iter)
        __syncthreads();      // all waves' async writes visible; prev readers done

        // Overlap: async-gather the NEXT tile into the other buffer.
        long nextTile = (long)tile + gridDim.x;
        if (nextTile < n_tiles) issue_gather(nextTile * TILE_M, buf ^ 1);

        // ---- WMMA: wave owns rows [wave*16, wave*16+16), all 128 cols ----
        const float* Acur = smem + buf * A_LDS_FLOATS;
        v8f acc[NTILES];
        #pragma unroll
        for (int nt = 0; nt < NTILES; ++nt) acc[nt] = (v8f){};

        const int arow  = wave * 16 + (lane & 15);
        const int abase = arow * A_STRIDE + ((lane >> 4) << 1);
        const float* bbase = &Wlds[lane * 2];

        #pragma unroll 4
        for (int ks = 0; ks < KSTEPS; ++ks) {
            v2f a = *(const v2f*)(&Acur[abase + ks * 4]);
            #pragma unroll
            for (int nt = 0; nt < NTILES; ++nt) {
                v2f b = *(const v2f*)(bbase + (ks * NTILES + nt) * 64);
                acc[nt] = __builtin_amdgcn_wmma_f32_16x16x4_f32(
                    false, a, false, b, (short)0, acc[nt], false, false);
            }
        }

        // ---- ReLU + NT store (C layout: lane<16 -> M=v, lane>=16 -> M=v+8) ----
        const long rowBase    = (long)tile * TILE_M;
        const int  colBase    = lane & 15;
        const long rStoreBase = rowBase + wave * 16 + ((lane >> 4) << 3);
        #pragma unroll
        for (int nt = 0; nt < NTILES; ++nt) {
            int col = nt * 16 + colBase;
            #pragma unroll
            for (int v = 0; v < 8; ++v) {
                long r = rStoreBase + v;
                if (r < n_rows) {
                    float x = acc[nt][v];
                    __builtin_nontemporal_store(x > 0.0f ? x : 0.0f,
                                                out + r * D_OUT_ + col);
                }
            }
        }
        buf ^= 1;
    }
}

extern "C" void kernel_launch(void* const* d_in, const int* in_sizes, int n_in,
                              void* d_out, int out_size, void* d_ws, size_t ws_size,
                              hipStream_t stream) {
    const float* review = (const float*)d_in[0];
    const float* user   = (const float*)d_in[1];
    const float* item   = (const float*)d_in[2];
    const float* W      = (const float*)d_in[3];
    const int*   adj_u  = (const int*)d_in[4];
    const int*   adj_i  = (const int*)d_in[5];
    float* out = (float*)d_out;

    int n_rows  = in_sizes[0] / D_ENT;
    int n_tiles = (n_rows + TILE_M - 1) / TILE_M;
    int blocks  = n_tiles < 2048 ? n_tiles : 2048;

    hipLaunchKernelGGL(concat_agg_gemm, dim3(blocks), dim3(256), LDS_BYTES, stream,
                       review, user, item, W, adj_u, adj_i, out, n_rows, n_tiles);
}